// GCNLayer_25228637896827
// MI455X (gfx1250) — compile-verified
//
#include <hip/hip_runtime.h>

typedef __attribute__((ext_vector_type(2))) float v2f;
typedef __attribute__((ext_vector_type(8))) float v8f;

#define N_NODES 40000
#define N_EDGES 640000
#define F       128   // F_IN == F_OUT == 128

// ---------------------------------------------------------------------------
// Kernel 1: xd = x * drop_mask * 2.0 (dropout with p=0.5), and zero agg.
// Vectorized 128-bit loads/stores (global_load_b128 / global_store_b128).
// ---------------------------------------------------------------------------
__global__ void gcn_prep_kernel(const float4* __restrict__ x,
                                const int4*   __restrict__ mask,
                                float4*       __restrict__ xd,
                                float4*       __restrict__ agg,
                                int total4) {
    int i = blockIdx.x * blockDim.x + threadIdx.x;
    if (i >= total4) return;
    float4 xv = x[i];
    int4   mv = mask[i];
    float4 r;
    r.x = xv.x * (float)mv.x * 2.0f;
    r.y = xv.y * (float)mv.y * 2.0f;
    r.z = xv.z * (float)mv.z * 2.0f;
    r.w = xv.w * (float)mv.w * 2.0f;
    xd[i] = r;
    agg[i] = make_float4(0.f, 0.f, 0.f, 0.f);
}

// ---------------------------------------------------------------------------
// Kernel 2: scatter-aggregate. One wave (32 lanes) per edge; each lane owns
// 4 consecutive features (float4 gather from xd, 4x global_atomic_add_f32
// into agg). xd (20.5 MB) and agg (20.5 MB) are L2-resident on MI455X.
// ---------------------------------------------------------------------------
__global__ void gcn_scatter_kernel(const float*  __restrict__ xd,
                                   const int*    __restrict__ erow,
                                   const int*    __restrict__ ecol,
                                   const float*  __restrict__ eval,
                                   float*        __restrict__ agg) {
    int gtid = blockIdx.x * blockDim.x + threadIdx.x;
    int edge = gtid >> 5;
    int lane = threadIdx.x & 31;
    if (edge >= N_EDGES) return;

    int   r = erow[edge];
    int   c = ecol[edge];
    float v = eval[edge];

    const float4* src = (const float4*)(xd + (size_t)c * F) + lane;
    float4 m = *src;

    float* dst = agg + (size_t)r * F + lane * 4;
    atomicAdd(dst + 0, v * m.x);
    atomicAdd(dst + 1, v * m.y);
    atomicAdd(dst + 2, v * m.z);
    atomicAdd(dst + 3, v * m.w);
}

// ---------------------------------------------------------------------------
// Kernel 3: out = agg @ W^T + b using V_WMMA_F32_16X16X4_F32 (exact fp32).
// Block = 256 threads = 8 waves; blockIdx.x = row tile (16 rows), wave id =
// column tile (16 cols) -> one block covers 16 x 128 of the output.
// K-loop: 128/4 = 32 WMMA ops per tile, accumulating in the 8-VGPR C frag.
//
// A layout (16x4 f32, ISA 7.12.2): lane L<16 holds row M=L, VGPR0/1 = K 0/1;
// lane L>=16 holds row M=L-16, VGPR0/1 = K 2/3. B (4x16) mirrored:
// lane gives N, half-wave selects K pair. W is [F_OUT][F_IN] row-major, so
// B[k][n] = W[n*128 + k] directly.
// ---------------------------------------------------------------------------
__global__ void gcn_gemm_wmma_kernel(const float* __restrict__ agg,
                                     const float* __restrict__ W,
                                     const float* __restrict__ bias,
                                     float*       __restrict__ out) {
    int lane   = threadIdx.x & 31;
    int waveId = threadIdx.x >> 5;          // 0..7 -> column tile
    int row0   = blockIdx.x * 16;           // 2500 row tiles
    int col0   = waveId * 16;

    int mn    = lane & 15;                  // M for A-frag, N for B-frag
    int khalf = (lane >> 4) * 2;            // 0 or 2

    const float* arow = agg + (size_t)(row0 + mn) * F;  // A: rows of agg
    const float* brow = W   + (size_t)(col0 + mn) * F;  // B: rows of W (= W^T cols)

    v8f c = {};
    #pragma unroll 8
    for (int kb = 0; kb < F; kb += 4) {
        int k = kb + khalf;
        v2f a; a.x = arow[k]; a.y = arow[k + 1];
        v2f b; b.x = brow[k]; b.y = brow[k + 1];
        // 8 args: (neg_a, A, neg_b, B, c_mod, C, reuse_a, reuse_b)
        c = __builtin_amdgcn_wmma_f32_16x16x4_f32(
                false, a, false, b, (short)0, c, false, false);
    }

    // C/D layout: VGPR v -> lanes 0-15: (M=v, N=lane); lanes 16-31: (M=v+8, N=lane-16)
    int Mbase = (lane >> 4) * 8;
    float bv = bias[col0 + mn];
    #pragma unroll
    for (int v = 0; v < 8; v++) {
        out[(size_t)(row0 + Mbase + v) * F + col0 + mn] = c[v] + bv;
    }
}

// ---------------------------------------------------------------------------
extern "C" void kernel_launch(void* const* d_in, const int* in_sizes, int n_in,
                              void* d_out, int out_size, void* d_ws, size_t ws_size,
                              hipStream_t stream) {
    // setup_inputs() order: x, edge_row, edge_col, edge_val, W, b, drop_mask
    const float* x    = (const float*)d_in[0];
    const int*   erow = (const int*)  d_in[1];
    const int*   ecol = (const int*)  d_in[2];
    const float* ev   = (const float*)d_in[3];
    const float* W    = (const float*)d_in[4];
    const float* b    = (const float*)d_in[5];
    const int*   mask = (const int*)  d_in[6];

    // workspace: xd [N,F] fp32 then agg [N,F] fp32  (2 * 20.48 MB)
    float* xd  = (float*)d_ws;
    float* agg = xd + (size_t)N_NODES * F;

    // 1) dropout-scale + zero agg
    int total4 = (N_NODES * F) / 4;   // 1,280,000 float4 elements
    gcn_prep_kernel<<<(total4 + 255) / 256, 256, 0, stream>>>(
        (const float4*)x, (const int4*)mask, (float4*)xd, (float4*)agg, total4);

    // 2) edge scatter: one wave per edge -> E*32 threads
    int scatter_blocks = (N_EDGES * 32) / 256;   // 80,000
    gcn_scatter_kernel<<<scatter_blocks, 256, 0, stream>>>(xd, erow, ecol, ev, agg);

    // 3) WMMA GEMM + bias -> d_out
    gcn_gemm_wmma_kernel<<<N_NODES / 16, 256, 0, stream>>>(agg, W, b, (float*)d_out);
}